// SSIM3D_DIST_85109071938057
// MI455X (gfx1250) — compile-verified
//
#include <hip/hip_runtime.h>
#include <math.h>

// ---------------------------------------------------------------------------
// SSIM-3D, fully fused single pass for MI455X (gfx1250, wave32, WMMA).
//
// Tile: 16x16x16 interior, 26^3 halo region of gt/pred in LDS.
// Halo load: GLOBAL_LOAD_ASYNC_TO_LDS_B32 (ASYNCcnt path, LDS pre-zeroed so
// EXEC-masked skips implement the conv SAME zero padding).
// Blur = banded-matrix GEMM on V_WMMA_F32_16X16X4_F32:
//   out16x16 = W0^T(16xK) * B0(Kx16) + W1^T * B1, K split into 4-wide chunks.
// Per-lane operand layouts follow CDNA5 ISA:
//   A (16x4 f32): lanes 0-15 hold rows M, K={0,1}; lanes 16-31 K={2,3}
//   B (4x16 f32): lane gives column N=lane&15, K pair = (lane>>4)*2 + {0,1}
//   C/D (16x16 f32): lane N=lane&15, VGPR r -> M = 8*(lane>>4)+r
// ---------------------------------------------------------------------------

typedef float v2f __attribute__((ext_vector_type(2)));
typedef float v4f __attribute__((ext_vector_type(4)));
typedef float v8f __attribute__((ext_vector_type(8)));

#define WMMA_F32(a, b, c) \
  __builtin_amdgcn_wmma_f32_16x16x4_f32(false, (a), false, (b), (short)0, (c), false, false)

// Gaussian band coefficient: weight of raw sample rx for output position m.
__device__ __forceinline__ float wcoef(const float* gsh, int m, int rx) {
  int d = rx - m;
  return (d >= 0 && d <= 10) ? gsh[d] : 0.0f;
}

// LDS float counts
#define RAW_N   21632   // 26*26*32  (x padded 26->32)
#define T1_N    13312   // 26(z) * 32(y pad) * 16(x)
#define T2_N     8192   // 32(z pad) * 16(x) * 16(y)
#define SMEM_N  (RAW_N * 2 + T1_N + T2_N)   // 64768 floats = 259072 B

__global__ __launch_bounds__(256, 1)
void ssim3d_tile_kernel(const float* __restrict__ gt,
                        const float* __restrict__ pr,
                        float* __restrict__ partial)
{
  extern __shared__ float smem[];
  float* sG = smem;                 // raw gt  [z26][y26][x32]
  float* sP = sG + RAW_N;           // raw pred
  float* t1 = sP + RAW_N;           // x-blur  [z26][ypad32][x16]
  float* t2 = t1 + T1_N;            // xy-blur [zpad32][x16][y16]
  __shared__ float gsh[11];
  __shared__ float red[256];

  const int tid  = threadIdx.x;
  const int lane = tid & 31;
  const int w    = tid >> 5;        // wave id, 0..7

  // Gaussian table in f64 (matches reference _gauss1d: float64 -> float32)
  if (tid == 0) {
    double tmp[11]; double s = 0.0;
    for (int i = 0; i < 11; ++i) {
      double x = (double)(i - 5);
      tmp[i] = exp(-(x * x) / 4.5);   // 2*sigma^2 = 4.5
      s += tmp[i];
    }
    for (int i = 0; i < 11; ++i) gsh[i] = (float)(tmp[i] / s);
  }

  // Zero all of LDS (pad regions must be 0 so zero-weight products in WMMA
  // never touch NaN garbage, and so skipped async loads leave zero padding).
  for (int i = tid; i < SMEM_N; i += 256) smem[i] = 0.0f;
  __syncthreads();

  // ---- tile decode: 12x12x12 tiles per volume, 4 volumes
  int t = blockIdx.x;
  const int tx = t % 12; t /= 12;
  const int ty = t % 12; t /= 12;
  const int tz = t % 12; t /= 12;
  const int nb = t;
  const long vol = 192L * 192L * 192L;
  const float* G = gt + (long)nb * vol;
  const float* P = pr + (long)nb * vol;
  const int ox = tx * 16 - 5, oy = ty * 16 - 5, oz = tz * 16 - 5;

  // ---- async halo load: 26^3 region, direct global -> LDS (ASYNCcnt path).
  // Out-of-volume lanes skip the op; pre-zeroed LDS provides SAME zero pad.
  for (int i = tid; i < 26 * 26 * 26; i += 256) {
    int iz = i / 676;
    int r  = i - iz * 676;
    int iy = r / 26;
    int ix = r - iy * 26;
    int gx = ox + ix, gy = oy + iy, gz = oz + iz;
    int so = (iz * 26 + iy) * 32 + ix;
    if ((unsigned)gx < 192u && (unsigned)gy < 192u && (unsigned)gz < 192u) {
      long off = ((long)gz * 192 + gy) * 192 + gx;
      unsigned ldsG = (unsigned)(uintptr_t)(sG + so);
      unsigned ldsP = (unsigned)(uintptr_t)(sP + so);
      const float* ga = G + off;
      const float* pa = P + off;
      asm volatile("global_load_async_to_lds_b32 %0, %1, off"
                   :: "v"(ldsG), "v"(ga) : "memory");
      asm volatile("global_load_async_to_lds_b32 %0, %1, off"
                   :: "v"(ldsP), "v"(pa) : "memory");
    }
  }
  asm volatile("s_wait_asynccnt 0x0" ::: "memory");
  __syncthreads();

  // ---- per-lane constant A operands (banded Gaussian, same for all stages)
  const int m  = lane & 15;          // A row (output position along blur axis)
  const int kb = (lane >> 4) * 2;    // K sub-pair within a 4-wide chunk
  const int n  = lane & 15;          // B column (perpendicular line index)
  v2f A0[4], A1[3];
#pragma unroll
  for (int c = 0; c < 4; ++c) {
    int k = c * 4 + kb;
    A0[c].x = wcoef(gsh, m, k);
    A0[c].y = wcoef(gsh, m, k + 1);
  }
#pragma unroll
  for (int c = 0; c < 3; ++c) {
    int k = 16 + c * 4 + kb;
    A1[c].x = wcoef(gsh, m, k);
    A1[c].y = wcoef(gsh, m, k + 1);
  }

  v8f acc[5][2];   // 5 blurred fields, 2 output blocks per wave

#pragma unroll
  for (int f = 0; f < 5; ++f) {
    // ================= stage 1: blur along x =================
    // 676 (y,z)-lines in 43 groups of 16; per group one 16x16 D block.
    for (int g = w; g < 43; g += 8) {
      int line0 = g * 16 + n;
      bool valid = (line0 <= 675);
      int line = valid ? line0 : 675;
      const int base = line * 32;
      v8f d = {0.f, 0.f, 0.f, 0.f, 0.f, 0.f, 0.f, 0.f};
#pragma unroll
      for (int c = 0; c < 4; ++c) {
        int k = c * 4 + kb;
        float g0 = sG[base + k], g1 = sG[base + k + 1];
        float p0 = sP[base + k], p1 = sP[base + k + 1];
        v2f b;
        if      (f == 0) { b.x = g0;      b.y = g1;      }
        else if (f == 1) { b.x = p0;      b.y = p1;      }
        else if (f == 2) { b.x = g0 * g0; b.y = g1 * g1; }
        else if (f == 3) { b.x = p0 * p0; b.y = p1 * p1; }
        else             { b.x = g0 * p0; b.y = g1 * p1; }
        d = WMMA_F32(A0[c], b, d);
      }
#pragma unroll
      for (int c = 0; c < 3; ++c) {
        int k = 16 + c * 4 + kb;
        float g0 = sG[base + k], g1 = sG[base + k + 1];
        float p0 = sP[base + k], p1 = sP[base + k + 1];
        v2f b;
        if      (f == 0) { b.x = g0;      b.y = g1;      }
        else if (f == 1) { b.x = p0;      b.y = p1;      }
        else if (f == 2) { b.x = g0 * g0; b.y = g1 * g1; }
        else if (f == 3) { b.x = p0 * p0; b.y = p1 * p1; }
        else             { b.x = g0 * p0; b.y = g1 * p1; }
        d = WMMA_F32(A1[c], b, d);
      }
      if (valid) {
        int y = line % 26, z = line / 26;
        float* o = t1 + (z * 32 + y) * 16 + (lane >> 4) * 8;
        *(v4f*)(o)     = (v4f){d[0], d[1], d[2], d[3]};
        *(v4f*)(o + 4) = (v4f){d[4], d[5], d[6], d[7]};
      }
    }
    __syncthreads();

    // ================= stage 2: blur along y =================
    // one 16x16 block per z-slice (26 slices). M=y_out, N=x, K=raw y.
    for (int z = w; z < 26; z += 8) {
      const float* src = t1 + z * 512;         // [ypad32][x16]
      v8f d = {0.f, 0.f, 0.f, 0.f, 0.f, 0.f, 0.f, 0.f};
#pragma unroll
      for (int c = 0; c < 4; ++c) {
        int k = c * 4 + kb;
        v2f b; b.x = src[k * 16 + n]; b.y = src[(k + 1) * 16 + n];
        d = WMMA_F32(A0[c], b, d);
      }
#pragma unroll
      for (int c = 0; c < 3; ++c) {
        int k = 16 + c * 4 + kb;
        v2f b; b.x = src[k * 16 + n]; b.y = src[(k + 1) * 16 + n];
        d = WMMA_F32(A1[c], b, d);
      }
      // t2 layout [z][x][y]: addr = z*256 + x*16 + y; x = N (lane), y = M (reg)
      float* o = t2 + z * 256 + n * 16 + (lane >> 4) * 8;
      *(v4f*)(o)     = (v4f){d[0], d[1], d[2], d[3]};
      *(v4f*)(o + 4) = (v4f){d[4], d[5], d[6], d[7]};
    }
    __syncthreads();

    // ================= stage 3: blur along z -> registers =================
    // 256 (x,y) lines in 16 groups; wave w owns groups w and w+8.
#pragma unroll
    for (int bIdx = 0; bIdx < 2; ++bIdx) {
      const int gr = w + 8 * bIdx;
      const float* src = t2 + gr * 16 + n;     // + k*256 walks raw z
      v8f d = {0.f, 0.f, 0.f, 0.f, 0.f, 0.f, 0.f, 0.f};
#pragma unroll
      for (int c = 0; c < 4; ++c) {
        int k = c * 4 + kb;
        v2f b; b.x = src[k * 256]; b.y = src[(k + 1) * 256];
        d = WMMA_F32(A0[c], b, d);
      }
#pragma unroll
      for (int c = 0; c < 3; ++c) {
        int k = 16 + c * 4 + kb;
        v2f b; b.x = src[k * 256]; b.y = src[(k + 1) * 256];
        d = WMMA_F32(A1[c], b, d);
      }
      acc[f][bIdx] = d;
    }
    // NOTE: no barrier needed here. Stage 3 reads t2; the next field's
    // stage 1 writes only t1, and its trailing barrier orders stage 3's t2
    // reads before stage 2's t2 writes of the next field.
  }

  // ---- SSIM map + deterministic reduction --------------------------------
  // (lane, bIdx, r) -> voxel: x = w+8*bIdx, y = lane&15, z = 8*(lane>>4)+r.
  // Mapping is identical for all 5 fields, so per-register math is exact.
  const float C1 = 1.0e-4f;   // (0.01*1)^2
  const float C2 = 9.0e-4f;   // (0.03*1)^2
  float ssum = 0.0f;
#pragma unroll
  for (int bIdx = 0; bIdx < 2; ++bIdx) {
#pragma unroll
    for (int r = 0; r < 8; ++r) {
      float mux = acc[0][bIdx][r];
      float muy = acc[1][bIdx][r];
      float xx  = acc[2][bIdx][r];
      float yy  = acc[3][bIdx][r];
      float xy  = acc[4][bIdx][r];
      float mux2 = mux * mux;
      float muy2 = muy * muy;
      float muxy = mux * muy;
      float sx  = xx - mux2;
      float sy  = yy - muy2;
      float sxy = xy - muxy;
      float num = (2.0f * muxy + C1) * (2.0f * sxy + C2);
      float den = (mux2 + muy2 + C1) * (sx + sy + C2);
      ssum += 1.0f - num / den;
    }
  }
  __syncthreads();      // all waves done reading t2 / smem before reuse
  red[tid] = ssum;
  __syncthreads();
#pragma unroll
  for (int s = 128; s > 0; s >>= 1) {
    if (tid < s) red[tid] += red[tid + s];
    __syncthreads();
  }
  if (tid == 0) partial[blockIdx.x] = red[0];
}

__global__ __launch_bounds__(256)
void ssim3d_finish_kernel(const float* __restrict__ partial,
                          float* __restrict__ out, int npart)
{
  __shared__ double red[256];
  double s = 0.0;
  for (int i = threadIdx.x; i < npart; i += 256) s += (double)partial[i];
  red[threadIdx.x] = s;
  __syncthreads();
#pragma unroll
  for (int st = 128; st > 0; st >>= 1) {
    if (threadIdx.x < st) red[threadIdx.x] += red[threadIdx.x + st];
    __syncthreads();
  }
  if (threadIdx.x == 0) out[0] = (float)(red[0] / 28311552.0);  // 4*192^3
}

extern "C" void kernel_launch(void* const* d_in, const int* in_sizes, int n_in,
                              void* d_out, int out_size, void* d_ws, size_t ws_size,
                              hipStream_t stream) {
  const float* gt = (const float*)d_in[0];
  const float* pr = (const float*)d_in[1];
  float* partial  = (float*)d_ws;       // 6912 floats = 27.6 KB
  float* out      = (float*)d_out;

  const int nblocks = 12 * 12 * 12 * 4; // 6912 tiles
  const size_t shmem = (size_t)SMEM_N * sizeof(float);  // 259072 B < 320 KB

  // Allow >64KB dynamic LDS (gfx1250 WGP has 320 KB).
  (void)hipFuncSetAttribute((const void*)ssim3d_tile_kernel,
                            hipFuncAttributeMaxDynamicSharedMemorySize,
                            (int)shmem);

  ssim3d_tile_kernel<<<nblocks, 256, shmem, stream>>>(gt, pr, partial);
  ssim3d_finish_kernel<<<1, 256, 0, stream>>>(partial, out, nblocks);
}